// GMM_30408368456042
// MI455X (gfx1250) — compile-verified
//
#include <hip/hip_runtime.h>
#include <math.h>

#define KCOMP 64
#define DM 256
#define NPTS 8192
#define TAUF 1.0f

typedef float v2f __attribute__((ext_vector_type(2)));
typedef float v8f __attribute__((ext_vector_type(8)));
typedef int   v4i __attribute__((ext_vector_type(4)));

// ---- gfx1250 async global->LDS path (guarded so either toolchain compiles) ----
#if __has_builtin(__builtin_amdgcn_global_load_async_to_lds_b128)
#define HAVE_ASYNC_LDS 1
#define GLB_CAST(p) ((__attribute__((address_space(1))) v4i*)(p))
#define LDS_CAST(p) ((__attribute__((address_space(3))) v4i*)(p))
#define ASYNC_CP_B128(gsrc, ldst) \
  __builtin_amdgcn_global_load_async_to_lds_b128(GLB_CAST(gsrc), LDS_CAST(ldst), 0, 0)
#else
#define HAVE_ASYNC_LDS 0
#endif

#if __has_builtin(__builtin_amdgcn_s_wait_asynccnt)
#define ASYNC_WAIT() __builtin_amdgcn_s_wait_asynccnt(0)
#else
#define ASYNC_WAIT() asm volatile("s_wait_asynccnt 0x0" ::: "memory")
#endif

// ---------------------------------------------------------------------------
// Kernel A: Sigma_k = A_k A_k^T + diag(softplus(Dvec_k))
// grid (16 row-strips, 64 k), 256 threads. LDS-tiled dot products.
// ---------------------------------------------------------------------------
__global__ __launch_bounds__(256)
void gmm_sigma_kernel(const float* __restrict__ A, const float* __restrict__ Dvec,
                      float* __restrict__ SigmaG)
{
  __shared__ __align__(16) float Ar[16 * 257];
  __shared__ __align__(16) float Ac[16 * 257];
  const int k   = blockIdx.y;
  const int s   = blockIdx.x;
  const int tid = threadIdx.x;
  const float* Ak = A + (size_t)k * (DM * DM);

  for (int r = 0; r < 16; ++r)
    Ar[r * 257 + tid] = Ak[(s * 16 + r) * DM + tid];

  const int i = tid >> 4;   // row within strip
  const int j = tid & 15;   // col within 16-block
  for (int jb = 0; jb < 16; ++jb) {
    __syncthreads();
    for (int r = 0; r < 16; ++r)
      Ac[r * 257 + tid] = Ak[(jb * 16 + r) * DM + tid];
    __syncthreads();
    float dot = 0.f;
    for (int c = 0; c < DM; ++c)
      dot = fmaf(Ar[i * 257 + c], Ac[j * 257 + c], dot);
    const int gi = s * 16 + i, gj = jb * 16 + j;
    float add = 0.f;
    if (gi == gj) {
      float dv = Dvec[k * DM + gi];
      add = (dv > 20.f) ? dv : log1pf(__expf(dv));   // softplus
    }
    SigmaG[(size_t)k * (DM * DM) + (size_t)gi * DM + gj] = dot + add;
  }
}

// ---------------------------------------------------------------------------
// Kernel B: in-place Cholesky of Sigma_k (lower), one workgroup per k.
// Right-looking rank-1 updates on global scratch (L2-resident).
// Emits invdiag and logdet.
// ---------------------------------------------------------------------------
__global__ __launch_bounds__(256)
void gmm_chol_kernel(float* __restrict__ Sg, float* __restrict__ invdG,
                     float* __restrict__ logdetG)
{
  __shared__ float colj[256];
  __shared__ float diag;
  const int k   = blockIdx.x;
  const int tid = threadIdx.x;
  float* S = Sg + (size_t)k * (DM * DM);

  for (int j = 0; j < DM; ++j) {
    if (tid == 0) {
      float d = sqrtf(S[j * DM + j]);
      S[j * DM + j] = d;
      diag = d;
    }
    __syncthreads();
    const float inv = 1.0f / diag;
    const int i = j + 1 + tid;
    float lij = 0.f;
    if (i < DM) {
      lij = S[i * DM + j] * inv;
      S[i * DM + j] = lij;
    }
    colj[tid] = lij;            // colj[t] = L[j+1+t][j]
    __threadfence();
    __syncthreads();
    if (i < DM) {
      for (int p = j + 1; p <= i; ++p)
        S[i * DM + p] -= lij * colj[p - j - 1];
    }
    __threadfence();
    __syncthreads();
  }

  invdG[k * DM + tid] = 1.0f / S[tid * DM + tid];
  if (tid == 0) {
    float ld = 0.f;
    for (int t = 0; t < DM; ++t) ld += logf(S[t * DM + t]);
    logdetG[k] = 2.0f * ld;
  }
}

// ---------------------------------------------------------------------------
// Kernel C: fused blocked forward substitution + Mahalanobis, WMMA f32.
// grid (NPTS/32 tiles, 64 k), 64 threads = 2 waves. Each wave owns 16 columns.
// For block-row I: U = L[I, 0:16I] @ Y  (chain of v_wmma_f32_16x16x4_f32),
// rhs = (x - mu) - U, 16x16 triangular solve per column, accumulate maha.
// L panel is staged with async global->LDS B128 copies (ASYNCcnt).
// ---------------------------------------------------------------------------
__global__ __launch_bounds__(64)
void gmm_solve_kernel(const float* __restrict__ x, const float* __restrict__ mu,
                      const float* __restrict__ Lg, const float* __restrict__ invdG,
                      const float* __restrict__ logdetG, float* __restrict__ logpG)
{
  __shared__ __align__(16) float Ysh[256 * 40]; // y, stride 40 (2*40 % 64 == 16: disjoint half-wave banks)
  __shared__ __align__(16) float Lp [16 * 260]; // L panel, stride 260 (== 4 mod 64: lanes spread)
  __shared__ __align__(16) float LII[16 * 20];  // diagonal block, 80B rows (B128-aligned)
  __shared__ __align__(16) float rhs[16 * 40];
  __shared__ __align__(16) float invd[16];
  __shared__ __align__(16) float muS[256];
  __shared__ __align__(16) float mahaS[32];

  const int k    = blockIdx.y;
  const int n0g  = blockIdx.x * 32;        // global column base of this tile
  const int tid  = threadIdx.x;            // 0..63
  const int lane = tid & 31;
  const int wave = tid >> 5;               // 0..1
  const int m16  = lane & 15;
  const int klo  = (lane >> 4) << 1;       // 0 or 2 (A/B half-wave K offset)
  const int rowoff = (lane >> 4) << 3;     // 0 or 8 (C/D half-wave M offset)
  const int wn0  = wave * 16;              // wave's column base within tile

  const float* Lk = Lg + (size_t)k * (DM * DM);

#if HAVE_ASYNC_LDS
  ASYNC_CP_B128(mu + k * DM + 4 * tid, &muS[4 * tid]);   // 64 x b128 = 256 floats
#else
  for (int c = tid; c < DM; c += 64) muS[c] = mu[k * DM + c];
#endif
  if (tid < 32) mahaS[tid] = 0.f;
#if HAVE_ASYNC_LDS
  ASYNC_WAIT();
#endif
  __syncthreads();

  for (int I = 0; I < 16; ++I) {
    const int C = I * 16;   // K-extent of the off-diagonal panel

#if HAVE_ASYNC_LDS
    // async-stage L panel rows 16I..16I+15, cols 0..C-1 in b128 chunks
    {
      const int c4max = C >> 2;
      for (int r = 0; r < 16; ++r) {
        const float* src = Lk + (size_t)(16 * I + r) * DM;
        for (int c4 = tid; c4 < c4max; c4 += 64)
          ASYNC_CP_B128(src + 4 * c4, &Lp[r * 260 + 4 * c4]);
      }
      // diagonal block: 64 b128 chunks, exactly one per thread
      const int r = tid >> 2, c4 = tid & 3;
      ASYNC_CP_B128(Lk + (size_t)(16 * I + r) * DM + C + 4 * c4,
                    &LII[r * 20 + 4 * c4]);
    }
    if (tid < 16) invd[tid] = invdG[k * DM + C + tid];
    ASYNC_WAIT();
#else
    for (int r = 0; r < 16; ++r) {
      const float* src = Lk + (size_t)(16 * I + r) * DM;
      for (int c = tid; c < C; c += 64) Lp[r * 260 + c] = src[c];
    }
    for (int idx = tid; idx < 256; idx += 64) {
      int r = idx >> 4, c = idx & 15;
      LII[r * 20 + c] = Lk[(size_t)(16 * I + r) * DM + (C + c)];
    }
    if (tid < 16) invd[tid] = invdG[k * DM + C + tid];
#endif
    __syncthreads();

    const int gn = n0g + wn0 + m16;
    if (I < 15)  // prefetch next block-row's x chunk (64B) for this column
      __builtin_prefetch(x + (size_t)gn * DM + 16 * (I + 1), 0, 3);

    // U = L[I, 0:C] @ Y[0:C, wave cols]   (two accumulators break the RAW chain)
    v8f acc0 = {};
    v8f acc1 = {};
    for (int c = 0; c < C; c += 8) {
      v2f a0, b0, a1, b1;
      a0.x = Lp[m16 * 260 + c + klo];
      a0.y = Lp[m16 * 260 + c + klo + 1];
      b0.x = Ysh[(c + klo) * 40 + wn0 + m16];
      b0.y = Ysh[(c + klo + 1) * 40 + wn0 + m16];
      acc0 = __builtin_amdgcn_wmma_f32_16x16x4_f32(false, a0, false, b0,
                                                   (short)0, acc0, false, false);
      a1.x = Lp[m16 * 260 + c + 4 + klo];
      a1.y = Lp[m16 * 260 + c + 4 + klo + 1];
      b1.x = Ysh[(c + 4 + klo) * 40 + wn0 + m16];
      b1.y = Ysh[(c + 4 + klo + 1) * 40 + wn0 + m16];
      acc1 = __builtin_amdgcn_wmma_f32_16x16x4_f32(false, a1, false, b1,
                                                   (short)0, acc1, false, false);
    }

    // rhs = (x - mu) - U ; C/D layout: row = r + rowoff, col = wn0 + m16
#pragma unroll
    for (int r = 0; r < 8; ++r) {
      const int M   = r + rowoff;
      const int row = 16 * I + M;
      const float diff = x[(size_t)gn * DM + row] - muS[row];
      rhs[M * 40 + wn0 + m16] = diff - (acc0[r] + acc1[r]);
    }
    __syncthreads();

    // 16-step forward substitution, one thread per column
    if (tid < 32) {
      const int n = tid;
      float y[16];
      float msum = 0.f;
#pragma unroll
      for (int i = 0; i < 16; ++i) {
        float sv = rhs[i * 40 + n];
        for (int p = 0; p < i; ++p) sv -= LII[i * 20 + p] * y[p];
        const float yi = sv * invd[i];
        y[i] = yi;
        Ysh[(16 * I + i) * 40 + n] = yi;
        msum += yi * yi;
      }
      mahaS[n] += msum;
    }
    __syncthreads();
  }

  if (tid < 32) {
    const float log2pi = 1.8378770664093453f;
    const float lp = -0.5f * (256.0f * log2pi + logdetG[k] + mahaS[tid]);
    logpG[(size_t)k * NPTS + n0g + tid] = lp;
  }
}

// ---------------------------------------------------------------------------
// Kernel D: mix_p = log_softmax(tau*pi) + logp; argmax (first max) +
// online logsumexp per point.
// ---------------------------------------------------------------------------
__global__ __launch_bounds__(256)
void gmm_mix_kernel(const float* __restrict__ pi, const float* __restrict__ logpG,
                    float* __restrict__ outAssign, float* __restrict__ nllG)
{
  const int n = blockIdx.x * 256 + threadIdx.x;
  if (n >= NPTS) return;

  float pm = -INFINITY;
  for (int k = 0; k < KCOMP; ++k) pm = fmaxf(pm, TAUF * pi[k]);
  float ps = 0.f;
  for (int k = 0; k < KCOMP; ++k) ps += __expf(TAUF * pi[k] - pm);
  const float lse_pi = pm + logf(ps);

  float bestv = -INFINITY;
  int   bestk = 0;
  float mm = -INFINITY, ss = 0.f;
  for (int k = 0; k < KCOMP; ++k) {
    const float v = TAUF * pi[k] - lse_pi + logpG[(size_t)k * NPTS + n];
    if (v > bestv) { bestv = v; bestk = k; }
    if (v > mm) { ss = ss * __expf(mm - v) + 1.f; mm = v; }
    else        { ss += __expf(v - mm); }
  }
  nllG[n]      = -(mm + logf(ss));
  outAssign[n] = (float)bestk;
}

// ---------------------------------------------------------------------------
// Kernel E: deterministic fixed-order mean of nll -> d_out[0]
// ---------------------------------------------------------------------------
__global__ __launch_bounds__(256)
void gmm_reduce_kernel(const float* __restrict__ nllG, float* __restrict__ out0)
{
  __shared__ float red[256];
  const int tid = threadIdx.x;
  float s = 0.f;
  for (int j = tid; j < NPTS; j += 256) s += nllG[j];
  red[tid] = s;
  __syncthreads();
  for (int w = 128; w > 0; w >>= 1) {
    if (tid < w) red[tid] += red[tid + w];
    __syncthreads();
  }
  if (tid == 0) out0[0] = red[0] / (float)NPTS;
}

// ---------------------------------------------------------------------------
extern "C" void kernel_launch(void* const* d_in, const int* in_sizes, int n_in,
                              void* d_out, int out_size, void* d_ws, size_t ws_size,
                              hipStream_t stream)
{
  (void)in_sizes; (void)n_in; (void)out_size; (void)ws_size;
  const float* x    = (const float*)d_in[0];   // (N, Dm)
  const float* A    = (const float*)d_in[1];   // (K, Dm, Dm)
  const float* Dvec = (const float*)d_in[2];   // (K, Dm)
  const float* mu   = (const float*)d_in[3];   // (K, Dm)
  const float* pi   = (const float*)d_in[4];   // (K,)
  float* out = (float*)d_out;                  // [0]=nll mean, [1..N]=assignment

  char* ws = (char*)d_ws;
  float* SigmaG  = (float*)(ws);                                     // 64*256*256 f32 (becomes L in place)
  float* invdG   = (float*)(ws + 16777216);                          // 64*256
  float* logdetG = (float*)(ws + 16777216 + 65536);                  // 64
  float* logpG   = (float*)(ws + 16777216 + 65536 + 256);            // 64*8192
  float* nllG    = (float*)(ws + 16777216 + 65536 + 256 + 2097152);  // 8192

  gmm_sigma_kernel <<<dim3(16, KCOMP),      256, 0, stream>>>(A, Dvec, SigmaG);
  gmm_chol_kernel  <<<dim3(KCOMP),          256, 0, stream>>>(SigmaG, invdG, logdetG);
  gmm_solve_kernel <<<dim3(NPTS/32, KCOMP),  64, 0, stream>>>(x, mu, SigmaG, invdG,
                                                              logdetG, logpG);
  gmm_mix_kernel   <<<dim3(NPTS/256),       256, 0, stream>>>(pi, logpG, out + 1, nllG);
  gmm_reduce_kernel<<<dim3(1),              256, 0, stream>>>(nllG, out);
}